// WideResNet_61641370632802
// MI455X (gfx1250) — compile-verified
//
#include <hip/hip_runtime.h>
#include <math.h>

#define B8    8
#define C16   16
#define K3    3
#define P4    4
#define STR   2
#define ITERS 3
#define PS    16          // P4*P4
#define KKB   72          // K3*K3*B8
#define EPSF  1e-8f
#define LAMF  1e-3f
#define LN2PI_F 1.8378770664093453f
#define NPW   4           // positions per workgroup

typedef __attribute__((ext_vector_type(2))) float v2f;
typedef __attribute__((ext_vector_type(8))) float v8f;

// x: (32,16,16,136) f32 | weights: (72,16,4,4) f32 | beta_u,beta_a: (16) f32
// out: (32,7,7,272) f32
__global__ __launch_bounds__(256)
void capsule_em_wmma(const float* __restrict__ x,
                     const float* __restrict__ w,
                     const float* __restrict__ beta_u,
                     const float* __restrict__ beta_a,
                     float* __restrict__ out)
{
    extern __shared__ float smem[];
    float* v    = smem;                      // [4][72][16][16]  73728
    float* r    = v    + NPW*KKB*C16*PS;     // [4][72][16]       4608
    float* ain  = r    + NPW*KKB*C16;        // [4][72]            288
    float* mu   = ain  + NPW*KKB;            // [4][16][16]       1024
    float* sig  = mu   + NPW*C16*PS;         // [4][16][16]       1024
    float* rsum = sig  + NPW*C16*PS;         // [4][16]             64
    float* aout = rsum + NPW*C16;            // [4][16]             64

    const int tid = threadIdx.x;
    const int wg  = blockIdx.x;              // 392 workgroups, 4 positions each

    // ---- Phase 0: activations a_in + init r = 1/C ----
    for (int idx = tid; idx < NPW*KKB; idx += 256) {
        int ns = idx / KKB, i = idx % KKB;
        int ky = i / (K3*B8), kx = (i / B8) % K3, bb = i % B8;
        int n  = wg*NPW + ns;
        int bi = n / 49, rem = n % 49;
        int oy = rem / 7, ox = rem % 7;
        int yy = oy*STR + ky, xx = ox*STR + kx;
        ain[idx] = x[((bi*16 + yy)*16 + xx)*136 + 128 + bb];
    }
    for (int idx = tid; idx < NPW*KKB*C16; idx += 256) r[idx] = 1.0f / C16;

    // ---- Phase 1: votes v[n,i,c] = p[n,i] @ W[i,c] via V_WMMA_F32_16X16X4_F32
    // A (16x4): rows M = 4 positions x 4 pose-rows, K = contraction b (4)
    // B (4x16): cols N = 4 capsules (of a group) x 4 pose-cols d
    // i-outer / cg-inner: one A-fragment load feeds 4 back-to-back WMMAs.
    const int wave = tid >> 5;
    const int lane = tid & 31;
    const int half = lane >> 4;              // 0: K={0,1}, 1: K={2,3}
    const int mrow = lane & 15;              // M for A rows / N for B cols
    const int b0   = half ? 2 : 0;
    const int ns_a = mrow >> 2, a_row = mrow & 3;     // A decomposition
    const int c_sb = mrow >> 2, d_col = mrow & 3;     // B decomposition
    {
        int n  = wg*NPW + ns_a;
        int bi = n / 49, rem = n % 49;
        int oy = rem / 7, ox = rem % 7;

        for (int i = wave; i < KKB; i += 8) {
            int ky = i / (K3*B8), kx = (i / B8) % K3, bb = i % B8;
            int yy = oy*STR + ky, xx = ox*STR + kx;

            // A fragment: p[n,i,a_row,b0..b0+1] (contiguous pair), invariant over cg
            const float* px = x + ((bi*16 + yy)*16 + xx)*136 + bb*PS + a_row*P4 + b0;
            v2f Af; Af.x = px[0]; Af.y = px[1];

            #pragma unroll
            for (int cg = 0; cg < 4; ++cg) {
                // B fragment: W[i, cg*4+c_sb, b0..b0+1, d_col]
                const float* pw = w + ((i*C16 + (cg*4 + c_sb))*P4 + b0)*P4 + d_col;
                v2f Bf; Bf.x = pw[0]; Bf.y = pw[P4];

                v8f acc = {};
                acc = __builtin_amdgcn_wmma_f32_16x16x4_f32(
                          false, Af, false, Bf, (short)0, acc, false, false);

                // D layout: vgpr k -> M = k (+8 for upper lanes), N = lane%16
                #pragma unroll
                for (int k = 0; k < 8; ++k) {
                    int M  = half ? (k + 8) : k;
                    int ns = M >> 2, aa = M & 3;
                    v[((ns*KKB + i)*C16 + (cg*4 + c_sb))*PS + aa*P4 + d_col] = acc[k];
                }
            }
        }
    }
    __syncthreads();

    // ---- Phase 2: EM routing on LDS-resident v ----
    for (int it = 0; it < ITERS; ++it) {
        // (a) rr = r*a_in, normalized over c (in-place; one thread per (n,i) row)
        for (int idx = tid; idx < NPW*KKB; idx += 256) {
            float a = ain[idx];
            float* rp = r + idx*C16;
            float s = 0.f;
            #pragma unroll
            for (int c = 0; c < C16; ++c) s += rp[c]*a;
            float g = a / (s + EPSF);
            #pragma unroll
            for (int c = 0; c < C16; ++c) rp[c] *= g;
        }
        __syncthreads();
        // (b) r_sum[c] = sum_i rr
        for (int idx = tid; idx < NPW*C16; idx += 256) {
            int ns = idx / C16, c = idx % C16;
            float s = 0.f;
            for (int i = 0; i < KKB; ++i) s += r[(ns*KKB + i)*C16 + c];
            rsum[idx] = s;
        }
        __syncthreads();
        // (c) mu
        for (int idx = tid; idx < NPW*C16*PS; idx += 256) {
            int ns = idx / (C16*PS), c = (idx / PS) % C16, ps = idx % PS;
            float s = 0.f;
            for (int i = 0; i < KKB; ++i)
                s += r[(ns*KKB + i)*C16 + c] * v[((ns*KKB + i)*C16 + c)*PS + ps];
            mu[idx] = s / (rsum[ns*C16 + c] + EPSF);
        }
        __syncthreads();
        // (d) sigma^2
        for (int idx = tid; idx < NPW*C16*PS; idx += 256) {
            int ns = idx / (C16*PS), c = (idx / PS) % C16, ps = idx % PS;
            float m = mu[idx], s = 0.f;
            for (int i = 0; i < KKB; ++i) {
                float dv = v[((ns*KKB + i)*C16 + c)*PS + ps] - m;
                s += r[(ns*KKB + i)*C16 + c] * dv * dv;
            }
            sig[idx] = s / (rsum[ns*C16 + c] + EPSF) + EPSF;
        }
        __syncthreads();
        // (e) a_out = sigmoid(lam*(beta_a - sum_ps cost_h))
        for (int idx = tid; idx < NPW*C16; idx += 256) {
            int c = idx % C16;
            float rs = rsum[idx], bu = beta_u[c], cost = 0.f;
            #pragma unroll
            for (int ps = 0; ps < PS; ++ps)
                cost += (bu + 0.5f*__logf(sig[idx*PS + ps])) * rs;
            float z = LAMF * (beta_a[c] - cost);
            aout[idx] = 1.f / (1.f + __expf(-z));
        }
        __syncthreads();
        // (f) r = softmax_c( sum_ps ln_p + log(a_out) )
        //     ln_ap staged in the thread-owned LDS r-row (no register array)
        if (it < ITERS - 1) {
            for (int idx = tid; idx < NPW*KKB; idx += 256) {
                int ns = idx / KKB;
                float* rp = r + idx*C16;
                float mx = -3.4e38f;
                for (int c = 0; c < C16; ++c) {
                    float s = 0.f;
                    #pragma unroll
                    for (int ps = 0; ps < PS; ++ps) {
                        float dv = v[(idx*C16 + c)*PS + ps] - mu[(ns*C16 + c)*PS + ps];
                        float sg = sig[(ns*C16 + c)*PS + ps];
                        s += -0.5f*dv*dv/sg - 0.5f*__logf(sg);
                    }
                    s += -0.5f*LN2PI_F*PS + __logf(aout[ns*C16 + c]);
                    rp[c] = s;
                    mx = fmaxf(mx, s);
                }
                float se = 0.f;
                for (int c = 0; c < C16; ++c) { float e = __expf(rp[c] - mx); rp[c] = e; se += e; }
                float inv = 1.f / se;
                for (int c = 0; c < C16; ++c) rp[c] *= inv;
            }
            __syncthreads();
        }
    }

    // ---- Phase 3: output (mu[256] ++ a_out[16]) per position ----
    for (int idx = tid; idx < NPW*272; idx += 256) {
        int ns = idx / 272, j = idx % 272;
        int n  = wg*NPW + ns;
        out[n*272 + j] = (j < 256) ? mu[ns*C16*PS + j] : aout[ns*C16 + (j - 256)];
    }
}

extern "C" void kernel_launch(void* const* d_in, const int* in_sizes, int n_in,
                              void* d_out, int out_size, void* d_ws, size_t ws_size,
                              hipStream_t stream) {
    (void)in_sizes; (void)n_in; (void)out_size; (void)d_ws; (void)ws_size;
    const float* x      = (const float*)d_in[0];
    const float* wts    = (const float*)d_in[1];
    const float* beta_u = (const float*)d_in[2];
    const float* beta_a = (const float*)d_in[3];
    float* out = (float*)d_out;

    // LDS: v(73728) + r(4608) + ain(288) + mu(1024) + sig(1024) + rsum(64) + aout(64)
    const size_t lds_bytes = (size_t)(73728 + 4608 + 288 + 1024 + 1024 + 64 + 64) * sizeof(float); // 323,200 B <= 320 KB/WGP
    const int n_positions = 32 * 7 * 7;        // 1568
    const int grid = n_positions / NPW;        // 392

    capsule_em_wmma<<<grid, 256, lds_bytes, stream>>>(x, wts, beta_u, beta_a, out);
}